// Seq2seq_75033078661747
// MI455X (gfx1250) — compile-verified
//
#include <hip/hip_runtime.h>
#include <math.h>

// ---------------- WMMA fp32 16x16x4 helpers (CDNA5) ----------------
typedef __attribute__((ext_vector_type(2))) float v2f;
typedef __attribute__((ext_vector_type(8))) float v8f;

__device__ __forceinline__ v8f wmma4(v2f a, v2f b, v8f c) {
  // D = A(16x4) * B(4x16) + C(16x16), all fp32
  return __builtin_amdgcn_wmma_f32_16x16x4_f32(false, a, false, b, (short)0, c,
                                               false, false);
}

__device__ __forceinline__ float sigmoidf_(float x) {
  return 1.f / (1.f + expf(-x));
}
__device__ __forceinline__ float warp_max_(float v) {
  for (int o = 16; o > 0; o >>= 1) v = fmaxf(v, __shfl_xor(v, o, 32));
  return v;
}
__device__ __forceinline__ float warp_sum_(float v) {
  for (int o = 16; o > 0; o >>= 1) v += __shfl_xor(v, o, 32);
  return v;
}

// Problem sizes
#define NB 32
#define NL 512
#define FIN 64
#define NEMB 256
#define NH 512
#define E2H 1024
#define G3H 1536
#define NT 64
#define NOUT 64

// ---------------- generic C = act(A @ W^T + bias), MB M-tiles per wave ----
// A: MxK row-major (stride lda), W: NxK row-major (stride ldw).
// Register blocking: the B (weight) fragment is loaded once per k-step and
// reused across MB WMMAs -> (MB+1)/MB VMEM loads per WMMA instead of 2.
template <int MB>
__global__ __launch_bounds__(256) void gemm_wmma(
    const float* __restrict__ A, long lda, const float* __restrict__ W, long ldw,
    const float* __restrict__ bias, float* __restrict__ C, long ldc,
    int M, int N, int K, int act) {
  int gw = (int)((blockIdx.x * (long)blockDim.x + threadIdx.x) >> 5);
  int lane = threadIdx.x & 31;
  int Nt = N >> 4, Mtb = (M >> 4) / MB;
  if (gw >= Mtb * Nt) return;  // wave-uniform -> EXEC stays all-ones for WMMA
  int mtb = gw / Nt, nt = gw % Nt;
  int n = (nt << 4) + (lane & 15);
  int kb = (lane >> 4) << 1;
  const float* wr = W + (long)n * ldw + kb;
  const float* ar[MB];
  v8f acc[MB];
#pragma unroll
  for (int i = 0; i < MB; ++i) {
    int m = ((mtb * MB + i) << 4) + (lane & 15);
    ar[i] = A + (long)m * lda + kb;
    acc[i] = {};
  }
  for (int k = 0; k < K; k += 4) {
    v2f b; b.x = wr[k]; b.y = wr[k + 1];
#pragma unroll
    for (int i = 0; i < MB; ++i) {
      v2f a; a.x = ar[i][k]; a.y = ar[i][k + 1];
      acc[i] = wmma4(a, b, acc[i]);
    }
  }
  float bv = bias ? bias[n] : 0.f;
#pragma unroll
  for (int i = 0; i < MB; ++i) {
    int mrb = ((mtb * MB + i) << 4) + ((lane >> 4) << 3);
    for (int r = 0; r < 8; ++r) {
      float v = acc[i][r] + bv;
      if (act == 1) v = fmaxf(v, 0.f);
      else if (act == 2) v = tanhf(v);
      C[(long)(mrb + r) * ldc + n] = v;
    }
  }
}

// Segmented-K variant: A = [A0 | A1 | A2] along K (boundaries ka1, ka2),
// W = [W0 | W1] along K (boundary kw1). MB=2 covers M=32 entirely, so the
// weight fragment is loaded exactly once per k-step. Optional 2nd output.
template <int MB>
__global__ __launch_bounds__(256) void gemm_cat_wmma(
    const float* __restrict__ A0, long lda0, const float* __restrict__ A1,
    long lda1, const float* __restrict__ A2, long lda2, int ka1, int ka2,
    const float* __restrict__ W0, long ldw0, const float* __restrict__ W1,
    long ldw1, int kw1, const float* __restrict__ bias0,
    const float* __restrict__ bias1, float* __restrict__ C, long ldc,
    float* __restrict__ C2, long ldc2, int M, int N, int K, int act) {
  int gw = (int)((blockIdx.x * (long)blockDim.x + threadIdx.x) >> 5);
  int lane = threadIdx.x & 31;
  int Nt = N >> 4, Mtb = (M >> 4) / MB;
  if (gw >= Mtb * Nt) return;
  int mtb = gw / Nt, nt = gw % Nt;
  int n = (nt << 4) + (lane & 15);
  int kb = (lane >> 4) << 1;
  int mrow[MB];
  v8f acc[MB];
#pragma unroll
  for (int i = 0; i < MB; ++i) {
    mrow[i] = ((mtb * MB + i) << 4) + (lane & 15);
    acc[i] = {};
  }
  for (int k = 0; k < K; k += 4) {  // k uniform -> segment select is SALU
    const float* ab; long als; int ao;
    if (k < ka1)      { ab = A0; als = lda0; ao = k; }
    else if (k < ka2) { ab = A1; als = lda1; ao = k - ka1; }
    else              { ab = A2; als = lda2; ao = k - ka2; }
    const float* wp; int wo;
    if (k < kw1) { wp = W0 + (long)n * ldw0; wo = k; }
    else         { wp = W1 + (long)n * ldw1; wo = k - kw1; }
    v2f b; b.x = wp[wo + kb]; b.y = wp[wo + kb + 1];
#pragma unroll
    for (int i = 0; i < MB; ++i) {
      const float* ap = ab + (long)mrow[i] * als + ao + kb;
      v2f a; a.x = ap[0]; a.y = ap[1];
      acc[i] = wmma4(a, b, acc[i]);
    }
  }
  float bv = (bias0 ? bias0[n] : 0.f) + (bias1 ? bias1[n] : 0.f);
#pragma unroll
  for (int i = 0; i < MB; ++i) {
    int mrb = ((mtb * MB + i) << 4) + ((lane >> 4) << 3);
    for (int r = 0; r < 8; ++r) {
      float v = acc[i][r] + bv;
      if (act == 1) v = fmaxf(v, 0.f);
      else if (act == 2) v = tanhf(v);
      C[(long)(mrb + r) * ldc + n] = v;
      if (C2) C2[(long)(mrb + r) * ldc2 + n] = v;
    }
  }
}

// ---------------- pos-emb add (pe indexed by batch, per reference) -------
__global__ void prep_kernel(const float* __restrict__ x, float* __restrict__ xp) {
  int i = blockIdx.x * blockDim.x + threadIdx.x;
  if (i >= NB * NL * FIN) return;
  int d = i & 63;
  int b = i >> 15;  // / (512*64)
  float ang = (float)b * expf((float)(d & ~1) * (-0.14391156831212787f)); // -ln(1e4)/64
  float pe = (d & 1) ? cosf(ang) : sinf(ang);
  xp[i] = x[i] + pe;
}

// ---------------- persistent bi-GRU scan -------------------------------
// grid = 2 blocks (dir 0 fwd, dir 1 bwd), 1024 threads = 32 waves.
// LDS: h (32x512 f32 = 64KB) + gh (32x1536 f32 = 192KB) = 256KB of the WGP's 320KB.
// Per step: gh = h @ Whh^T via fp32 WMMA. Each wave owns 3 N-tiles x 2 M-tiles;
// the global Whh fragment is loaded once per k-step and reused for both M-tiles
// (A fragments come from LDS). Then elementwise GRU gate combine; writes
// relu(h + res) in place into enc_out (fused residual + ReLU).
__global__ __launch_bounds__(1024) void gru_scan_kernel(
    const float* __restrict__ xgf, const float* __restrict__ xgb,
    const float* __restrict__ Whhf, const float* __restrict__ Whhb,
    const float* __restrict__ bhhf, const float* __restrict__ bhhb,
    float* __restrict__ enc, float* __restrict__ hcat) {
  extern __shared__ float lds[];
  float* hs = lds;              // 16384
  float* gs = lds + NB * NH;    // 49152
  const int dir = blockIdx.x;
  const float* xg = dir ? xgb : xgf;
  const float* Whh = dir ? Whhb : Whhf;
  const float* bhh = dir ? bhhb : bhhf;
  const int colbase = dir ? NH : 0;
  const int tid = threadIdx.x;
  const int wave = tid >> 5, lane = tid & 31;
  const int ln15 = lane & 15, kb = (lane >> 4) << 1;
  for (int i = tid; i < NB * NH; i += 1024) hs[i] = 0.f;
  __syncthreads();
  for (int s = 0; s < NL; ++s) {
    const int l = dir ? (NL - 1 - s) : s;
    // gh = h @ Whh^T  (M=32, N=1536, K=512): 96 N-tiles / 32 waves = 3 each
    for (int t3 = 0; t3 < 3; ++t3) {
      int nt = wave * 3 + t3;
      int n = (nt << 4) + ln15;
      const float* wr = Whh + (long)n * NH + kb;
      const float* h0 = hs + ln15 * NH + kb;         // M-tile 0: rows 0..15
      const float* h1 = hs + (16 + ln15) * NH + kb;  // M-tile 1: rows 16..31
      v8f acc0 = {}, acc1 = {};
      for (int k = 0; k < NH; k += 4) {
        v2f b; b.x = wr[k]; b.y = wr[k + 1];
        v2f a0; a0.x = h0[k]; a0.y = h0[k + 1];
        v2f a1; a1.x = h1[k]; a1.y = h1[k + 1];
        acc0 = wmma4(a0, b, acc0);
        acc1 = wmma4(a1, b, acc1);
      }
      int mrb = (lane >> 4) << 3;
      for (int r = 0; r < 8; ++r) {
        gs[(mrb + r) * G3H + n] = acc0[r];
        gs[(16 + mrb + r) * G3H + n] = acc1[r];
      }
    }
    __syncthreads();
    // gate combine: 16384 elements, 16 per thread
    for (int i = 0; i < 16; ++i) {
      int e = tid + (i << 10);
      int b = e >> 9, j = e & 511;
      const float* xr_ = xg + ((long)(b * NL + l)) * G3H;
      float xr = xr_[j], xz = xr_[NH + j], xn = xr_[2 * NH + j];
      float hr = gs[b * G3H + j] + bhh[j];
      float hz = gs[b * G3H + NH + j] + bhh[NH + j];
      float hn = gs[b * G3H + 2 * NH + j] + bhh[2 * NH + j];
      float rg = sigmoidf_(xr + hr);
      float zg = sigmoidf_(xz + hz);
      float ng = tanhf(xn + rg * hn);
      float hnew = (1.f - zg) * ng + zg * hs[e];
      hs[e] = hnew;
      long oi = ((long)(b * NL + l)) * E2H + colbase + j;
      enc[oi] = fmaxf(hnew + enc[oi], 0.f);  // fused residual + ReLU (in place)
    }
    __syncthreads();
  }
  for (int i = 0; i < 16; ++i) {
    int e = tid + (i << 10);
    int b = e >> 9, j = e & 511;
    hcat[b * E2H + colbase + j] = hs[e];
  }
}

// ---------------- attention: scores + softmax + weighted sum -------------
// grid = 32 (one block per batch), 512 threads.
__global__ __launch_bounds__(512) void attn_kernel(
    const float* __restrict__ eattn, const float* __restrict__ hwed,
    const float* __restrict__ vat, const float* __restrict__ enc,
    float* __restrict__ weighted) {
  __shared__ float hv[NH], sv[NH], sc[NL], red[16];
  int b = blockIdx.x, tid = threadIdx.x;
  hv[tid] = hwed[b * NH + tid];
  sv[tid] = vat[tid];
  __syncthreads();
  const float* ea = eattn + ((long)(b * NL + tid)) * NH;
  float s = 0.f;
  for (int j = 0; j < NH; ++j) s += sv[j] * tanhf(ea[j] + hv[j]);
  float wm = warp_max_(s);
  if ((tid & 31) == 0) red[tid >> 5] = wm;
  __syncthreads();
  float mx = red[0];
  for (int i = 1; i < 16; ++i) mx = fmaxf(mx, red[i]);
  __syncthreads();
  float ex = expf(s - mx);
  float wsum = warp_sum_(ex);
  if ((tid & 31) == 0) red[tid >> 5] = wsum;
  __syncthreads();
  float sum = 0.f;
  for (int i = 0; i < 16; ++i) sum += red[i];
  sc[tid] = ex / sum;
  __syncthreads();
  const float* eo = enc + (long)b * NL * E2H;
  for (int e = tid; e < E2H; e += 512) {
    float acc = 0.f;
    for (int l = 0; l < NL; ++l) acc += sc[l] * eo[(long)l * E2H + e];
    weighted[b * E2H + e] = acc;
  }
}

// ---------------- decoder GRU cell (gx, gh kept separate: n = tanh(xn + r*hn))
__global__ void dec_gru_cell(const float* __restrict__ gx,
                             const float* __restrict__ gh,
                             float* __restrict__ h) {
  int i = blockIdx.x * blockDim.x + threadIdx.x;  // 32*512
  int b = i >> 9, j = i & 511;
  const float* gxr = gx + (long)b * G3H;
  const float* ghr = gh + (long)b * G3H;
  float r = sigmoidf_(gxr[j] + ghr[j]);
  float z = sigmoidf_(gxr[NH + j] + ghr[NH + j]);
  float n = tanhf(gxr[2 * NH + j] + r * ghr[2 * NH + j]);
  h[i] = (1.f - z) * n + z * h[i];
}

// ---------------- workspace layout (floats) -----------------------------
static const size_t OFF_XP = 0;                          // 1,048,576
static const size_t OFF_EMB = OFF_XP + 1048576;          // 4,194,304
static const size_t OFF_ENC = OFF_EMB + 4194304;         // 16,777,216 (res -> enc_out)
static const size_t OFF_XGF = OFF_ENC + 16777216;        // 25,165,824
static const size_t OFF_XGB = OFF_XGF + 25165824;        // 25,165,824
static const size_t OFF_EATT = OFF_XGB + 25165824;       // 8,388,608
static const size_t OFF_HCAT = OFF_EATT + 8388608;       // 32,768
static const size_t OFF_H = OFF_HCAT + 32768;            // 16,384
static const size_t OFF_EMBT = OFF_H + 16384;            // 8,192
static const size_t OFF_WGT = OFF_EMBT + 8192;           // 32,768
static const size_t OFF_HWED = OFF_WGT + 32768;          // 16,384
static const size_t OFF_GX = OFF_HWED + 16384;           // 49,152
static const size_t OFF_GH = OFF_GX + 49152;             // 49,152
static const size_t OFF_DIN = OFF_GH + 49152;            // 2,048

static inline void launch_gemm(const float* A, long lda, const float* W,
                               long ldw, const float* bias, float* C, long ldc,
                               int M, int N, int K, int act, int mb,
                               hipStream_t s) {
  if (mb == 4) {
    int tiles = (M / 64) * (N / 16);
    hipLaunchKernelGGL(gemm_wmma<4>, dim3((tiles + 7) / 8), dim3(256), 0, s, A,
                       lda, W, ldw, bias, C, ldc, M, N, K, act);
  } else {
    int tiles = (M / 32) * (N / 16);
    hipLaunchKernelGGL(gemm_wmma<2>, dim3((tiles + 7) / 8), dim3(256), 0, s, A,
                       lda, W, ldw, bias, C, ldc, M, N, K, act);
  }
}

extern "C" void kernel_launch(void* const* d_in, const int* in_sizes, int n_in,
                              void* d_out, int out_size, void* d_ws,
                              size_t ws_size, hipStream_t stream) {
  (void)in_sizes; (void)n_in; (void)out_size; (void)ws_size;
  const float* x      = (const float*)d_in[0];
  const float* W_cast = (const float*)d_in[1];
  const float* b_cast = (const float*)d_in[2];
  const float* W_ee   = (const float*)d_in[3];
  const float* b_ee   = (const float*)d_in[4];
  const float* Wih_f  = (const float*)d_in[5];
  const float* Whh_f  = (const float*)d_in[6];
  const float* bih_f  = (const float*)d_in[7];
  const float* bhh_f  = (const float*)d_in[8];
  const float* Wih_b  = (const float*)d_in[9];
  const float* Whh_b  = (const float*)d_in[10];
  const float* bih_b  = (const float*)d_in[11];
  const float* bhh_b  = (const float*)d_in[12];
  const float* W_res  = (const float*)d_in[13];
  const float* b_res  = (const float*)d_in[14];
  const float* W_fc   = (const float*)d_in[15];
  const float* b_fc   = (const float*)d_in[16];
  const float* W_attn = (const float*)d_in[17];
  const float* b_attn = (const float*)d_in[18];
  const float* v_attn = (const float*)d_in[19];
  const float* W_de   = (const float*)d_in[20];
  const float* b_de   = (const float*)d_in[21];
  const float* Wdih   = (const float*)d_in[22];
  const float* Wdhh   = (const float*)d_in[23];
  const float* bdih   = (const float*)d_in[24];
  const float* bdhh   = (const float*)d_in[25];
  const float* W_out  = (const float*)d_in[26];
  const float* b_out  = (const float*)d_in[27];
  float* dout = (float*)d_out;

  float* ws = (float*)d_ws;
  float* xp = ws + OFF_XP;
  float* emb = ws + OFF_EMB;
  float* enc = ws + OFF_ENC;     // res, then enc_out in place
  float* xgf = ws + OFF_XGF;
  float* xgb = ws + OFF_XGB;
  float* eattn = ws + OFF_EATT;
  float* hcat = ws + OFF_HCAT;
  float* hbuf = ws + OFF_H;
  float* embt = ws + OFF_EMBT;
  float* wgt = ws + OFF_WGT;
  float* hwed = ws + OFF_HWED;
  float* gx = ws + OFF_GX;
  float* gh = ws + OFF_GH;
  float* din = ws + OFF_DIN;

  // allow 256KB dynamic LDS for the persistent scan (gfx1250 WGP has 320KB)
  (void)hipFuncSetAttribute((const void*)gru_scan_kernel,
                            hipFuncAttributeMaxDynamicSharedMemorySize, 262144);

  const int MBL = NB * NL;  // 16384

  // 1) x + pos_emb
  hipLaunchKernelGGL(prep_kernel, dim3(4096), dim3(256), 0, stream, x, xp);
  // 2) emb = relu(xp @ W_ee^T + b_ee)
  launch_gemm(xp, FIN, W_ee, FIN, b_ee, emb, NEMB, MBL, NEMB, FIN, 1, 4, stream);
  // 3) res = xp @ W_res^T + b_res   (into enc buffer, fused later in scan)
  launch_gemm(xp, FIN, W_res, FIN, b_res, enc, E2H, MBL, E2H, FIN, 0, 4, stream);
  // 4) dec_input = xp[:, L-1, :] @ W_cast^T + b_cast  (A row stride = L*FIN)
  launch_gemm(xp + (long)(NL - 1) * FIN, (long)NL * FIN, W_cast, FIN, b_cast,
              din, NOUT, NB, NOUT, FIN, 0, 2, stream);
  // 5) input gates for both GRU directions (hoisted out of the serial scan)
  launch_gemm(emb, NEMB, Wih_f, NEMB, bih_f, xgf, G3H, MBL, G3H, NEMB, 0, 4, stream);
  launch_gemm(emb, NEMB, Wih_b, NEMB, bih_b, xgb, G3H, MBL, G3H, NEMB, 0, 4, stream);
  // 6) persistent bi-GRU scan (2 WGPs), fused residual+ReLU -> enc_out
  hipLaunchKernelGGL(gru_scan_kernel, dim3(2), dim3(1024), 262144, stream, xgf,
                     xgb, Whh_f, Whh_b, bhh_f, bhh_b, enc, hcat);
  // 7) enc_attn = enc_out @ We_enc^T + b_attn  (We_enc = W_attn[:, :1024])
  launch_gemm(enc, E2H, W_attn, G3H, b_attn, eattn, NH, MBL, NH, E2H, 0, 4, stream);
  // 8) dec_hidden = tanh(hcat @ W_fc^T + b_fc)
  launch_gemm(hcat, E2H, W_fc, E2H, b_fc, hbuf, NH, NB, NH, E2H, 2, 2, stream);

  // 9) decoder loop (T=64); kernel boundaries act as device-wide barriers
  for (int t = 0; t < NT; ++t) {
    // embedded = dec_in @ W_de^T + b_de
    launch_gemm(din, NOUT, W_de, NOUT, b_de, embt, NEMB, NB, NEMB, NOUT, 0, 2, stream);
    // hWed = h @ We_dec^T   (We_dec = W_attn[:, 1024:1536])
    launch_gemm(hbuf, NH, W_attn + E2H, G3H, nullptr, hwed, NH, NB, NH, NH, 0, 2, stream);
    // attention: scores -> softmax over L -> weighted context
    hipLaunchKernelGGL(attn_kernel, dim3(NB), dim3(512), 0, stream, eattn, hwed,
                       v_attn, enc, wgt);
    // gx = [embedded | weighted] @ Wdih^T + bdih
    hipLaunchKernelGGL(gemm_cat_wmma<2>, dim3(12), dim3(256), 0, stream, embt,
                       (long)NEMB, wgt, (long)E2H, (const float*)nullptr, 0L,
                       NEMB, NEMB + E2H, Wdih, (long)(NEMB + E2H),
                       (const float*)nullptr, 0L, NEMB + E2H, bdih,
                       (const float*)nullptr, gx, (long)G3H, (float*)nullptr, 0L,
                       NB, G3H, NEMB + E2H, 0);
    // gh = h @ Wdhh^T + bdhh
    launch_gemm(hbuf, NH, Wdhh, NH, bdhh, gh, G3H, NB, G3H, NH, 0, 2, stream);
    // GRU cell -> h (in place)
    hipLaunchKernelGGL(dec_gru_cell, dim3(64), dim3(256), 0, stream, gx, gh, hbuf);
    // pred = [h_new | weighted | embedded] @ W_out^T + b_out
    //        -> next dec_in, and -> d_out[:, t, :]
    hipLaunchKernelGGL(gemm_cat_wmma<2>, dim3(1), dim3(256), 0, stream, hbuf,
                       (long)NH, wgt, (long)E2H, embt, (long)NEMB, NH, NH + E2H,
                       W_out, (long)(NH + E2H + NEMB), (const float*)nullptr, 0L,
                       NH + E2H + NEMB, b_out, (const float*)nullptr, din,
                       (long)NOUT, dout + (long)t * NOUT, (long)(NT * NOUT), NB,
                       NOUT, NH + E2H + NEMB, 0);
  }
}